// featMDITRE_72765335928867
// MI455X (gfx1250) — compile-verified
//
#include <hip/hip_runtime.h>

typedef __attribute__((ext_vector_type(2))) float v2f;
typedef __attribute__((ext_vector_type(8))) float v8f;

#define S_DIM 256
#define J_DIM 4000
#define R_DIM 256   // K*I = 8*32
#define D_DIM 128
#define EPSF  1e-10f

// -------------------------------------------------------------------------
// Kernel 1: squared row norms of eta (256 x 128) and dist (4000 x 128).
// One wave per row; 4 elements per lane; shfl_xor tree reduction.
// -------------------------------------------------------------------------
__global__ void norms_kernel(const float* __restrict__ eta,
                             const float* __restrict__ dist,
                             float* __restrict__ eta_n2,
                             float* __restrict__ dist_n2) {
    int b    = blockIdx.x;
    int lane = threadIdx.x;
    const float* row = (b < R_DIM) ? (eta + (size_t)b * D_DIM)
                                   : (dist + (size_t)(b - R_DIM) * D_DIM);
    float s = 0.0f;
#pragma unroll
    for (int i = 0; i < D_DIM; i += 32) {
        float v = row[lane + i];
        s += v * v;
    }
#pragma unroll
    for (int m = 16; m >= 1; m >>= 1) s += __shfl_xor(s, m, 32);
    if (lane == 0) {
        if (b < R_DIM) eta_n2[b] = s;
        else           dist_n2[b - R_DIM] = s;
    }
}

// -------------------------------------------------------------------------
// Kernel 2: W[r,j] = straight-through-hard( sigmoid((exp(kappa[r]) - d)*k_otu) )
// d = sqrt(|eta_r|^2 + |dist_j|^2 - 2*eta_r.dist_j), dot via WMMA f32 16x16x4.
// One wave computes a 16x16 (r,j) tile; K-loop over D=128 in steps of 4.
// A-frag: lane L, reg v -> eta[r0 + L%16][k + v + 2*(L/16)]
// B-frag: lane L, reg v -> dist[j0 + L%16][k + v + 2*(L/16)]  (B = dist^T)
// -------------------------------------------------------------------------
__global__ void wts_kernel(const float* __restrict__ eta,
                           const float* __restrict__ dist,
                           const float* __restrict__ kappa,
                           const float* __restrict__ eta_n2,
                           const float* __restrict__ dist_n2,
                           const int*   __restrict__ k_otu_p,
                           float* __restrict__ W) {
    int j0   = blockIdx.x * 16;
    int r0   = blockIdx.y * 16;
    int lane = threadIdx.x;
    int half = lane >> 4;        // 0 or 1
    int lm   = lane & 15;
    float k_otu = (float)k_otu_p[0];

    const float* ap = eta  + (size_t)(r0 + lm) * D_DIM + 2 * half;
    const float* bp = dist + (size_t)(j0 + lm) * D_DIM + 2 * half;

    v8f c = {};
#pragma unroll 8
    for (int k = 0; k < D_DIM; k += 4) {
        v2f a = *(const v2f*)(ap + k);
        v2f b = *(const v2f*)(bp + k);
        c = __builtin_amdgcn_wmma_f32_16x16x4_f32(
                /*neg_a=*/false, a, /*neg_b=*/false, b,
                /*c_mod=*/(short)0, c, /*reuse_a=*/false, /*reuse_b=*/false);
    }

    int   j  = j0 + lm;
    float dn = dist_n2[j];
#pragma unroll
    for (int v = 0; v < 8; ++v) {
        int   r    = r0 + v + 8 * half;
        float d2   = eta_n2[r] + dn - 2.0f * c[v];
        float d    = sqrtf(fmaxf(d2, 0.0f));
        float kap  = __expf(kappa[r]);
        float z    = (kap - d) * k_otu;
        float soft = 1.0f / (1.0f + __expf(-z));
        float hard = (soft > 0.5f) ? 1.0f : 0.0f;
        // straight-through forward value: (hard - soft) + soft
        W[(size_t)r * J_DIM + j] = (hard - soft) + soft;
    }
}

// -------------------------------------------------------------------------
// Kernel 3: den[r] = sum_j W[r,j]. One wave per row, shfl reduction.
// -------------------------------------------------------------------------
__global__ void den_kernel(const float* __restrict__ W,
                           float* __restrict__ den) {
    int r    = blockIdx.x;
    int lane = threadIdx.x;
    const float* row = W + (size_t)r * J_DIM;
    float s = 0.0f;
    for (int j = lane; j < J_DIM; j += 32) s += row[j];
#pragma unroll
    for (int m = 16; m >= 1; m >>= 1) s += __shfl_xor(s, m, 32);
    if (lane == 0) den[r] = s;
}

// -------------------------------------------------------------------------
// Kernel 4: num[s,r] = sum_j W[r,j] * x[s,j]  (WMMA, K = 4000)
// then out[s*256 + r] = sigmoid(((num+eps)/(den[r]+eps) - thresh[r])*k_thresh)
// A = W (256 x 4000), B = x^T (4000 x 256): B[k=j][n=s] = x[s][j].
// -------------------------------------------------------------------------
__global__ void out_kernel(const float* __restrict__ W,
                           const float* __restrict__ x,
                           const float* __restrict__ den,
                           const float* __restrict__ thresh,
                           const int*   __restrict__ k_thresh_p,
                           float* __restrict__ out) {
    int s0   = blockIdx.x * 16;
    int r0   = blockIdx.y * 16;
    int lane = threadIdx.x;
    int half = lane >> 4;
    int lm   = lane & 15;
    float k_th = (float)k_thresh_p[0];

    const float* ap = W + (size_t)(r0 + lm) * J_DIM + 2 * half;
    const float* bp = x + (size_t)(s0 + lm) * J_DIM + 2 * half;

    v8f c = {};
#pragma unroll 8
    for (int k = 0; k < J_DIM; k += 4) {
        v2f a = *(const v2f*)(ap + k);
        v2f b = *(const v2f*)(bp + k);
        c = __builtin_amdgcn_wmma_f32_16x16x4_f32(
                false, a, false, b, (short)0, c, false, false);
    }

    int s = s0 + lm;
#pragma unroll
    for (int v = 0; v < 8; ++v) {
        int   r   = r0 + v + 8 * half;
        float num = c[v] + EPSF;
        float xo  = num / (den[r] + EPSF);
        float z   = (xo - thresh[r]) * k_th;
        out[(size_t)s * R_DIM + r] = 1.0f / (1.0f + __expf(-z));
    }
}

// -------------------------------------------------------------------------
extern "C" void kernel_launch(void* const* d_in, const int* in_sizes, int n_in,
                              void* d_out, int out_size, void* d_ws, size_t ws_size,
                              hipStream_t stream) {
    (void)in_sizes; (void)n_in; (void)out_size; (void)ws_size;

    const float* x       = (const float*)d_in[0];  // [256, 4000, 1]
    const float* eta     = (const float*)d_in[1];  // [256, 128]
    const float* kappa   = (const float*)d_in[2];  // [8, 32] flat 256
    const float* thresh  = (const float*)d_in[3];  // [8, 32] flat 256
    const float* dist    = (const float*)d_in[4];  // [4000, 128]
    const int*   k_otu   = (const int*)d_in[5];    // scalar
    const int*   k_thr   = (const int*)d_in[6];    // scalar
    float*       out     = (float*)d_out;          // [256, 8, 32] flat

    // workspace layout
    float* W       = (float*)d_ws;                     // 256*4000 floats = 4.096 MB
    float* eta_n2  = W + (size_t)R_DIM * J_DIM;        // 256
    float* dist_n2 = eta_n2 + R_DIM;                   // 4000
    float* den     = dist_n2 + J_DIM;                  // 256

    norms_kernel<<<R_DIM + J_DIM, 32, 0, stream>>>(eta, dist, eta_n2, dist_n2);
    wts_kernel<<<dim3(J_DIM / 16, R_DIM / 16), 32, 0, stream>>>(
        eta, dist, kappa, eta_n2, dist_n2, k_otu, W);
    den_kernel<<<R_DIM, 32, 0, stream>>>(W, den);
    out_kernel<<<dim3(S_DIM / 16, R_DIM / 16), 32, 0, stream>>>(
        W, x, den, thresh, k_thr, out);
}